// MeshConv_69956427317463
// MI455X (gfx1250) — compile-verified
//
#include <hip/hip_runtime.h>

typedef float v2f __attribute__((ext_vector_type(2)));
typedef float v8f __attribute__((ext_vector_type(8)));

#define CIN        32
#define OUTF       64
#define KDIM       160      // 5*CIN
#define STR        164      // padded LDS row stride (floats): 164 mod 64 = 36 -> period-16 bank rotation
#define ROWS_PER_WG 128
#define NTHREADS   256

// LDS layout (dynamic): [ W : 64 x STR ][ combined : 128 x STR ]  = (64+128)*164*4 = 125,952 B
#define SMEM_FLOATS ((OUTF + ROWS_PER_WG) * STR)

__global__ __launch_bounds__(NTHREADS)
void meshconv_wmma_f32_kernel(const float* __restrict__ x,
                              const long long* __restrict__ nbr,
                              const float* __restrict__ W,
                              const float* __restrict__ bias,
                              float* __restrict__ out,
                              int E)
{
    extern __shared__ float smem[];
    float* Wl   = smem;                 // 64 rows x STR
    float* comb = smem + OUTF * STR;    // 128 rows x STR

    const int tid  = threadIdx.x;
    const int lane = tid & 31;
    const int wave = tid >> 5;
    const int base = blockIdx.x * ROWS_PER_WG;

    // ---- Stage 0: W (64x160, row-major) -> LDS with padded stride ----
    for (int i = tid; i < OUTF * KDIM; i += NTHREADS) {
        int r = i / KDIM;
        int c = i - r * KDIM;
        Wl[r * STR + c] = W[i];
    }

    // ---- Stage 1: build combined features (128 x 160) in LDS ----
    // lane = channel (coalesced 128B row loads); each wave builds 16 rows.
    const int ch = lane;
    for (int it = 0; it < 16; ++it) {
        const int el = wave + it * 8;       // 0..127
        const int e  = base + el;
        float* row = comb + el * STR;
        if (e < E) {
            float xv = x[(long long)e * CIN + ch];
            float g[4];
            #pragma unroll
            for (int j = 0; j < 4; ++j) {
                long long n = nbr[(long long)e * 4 + j];
                g[j] = (n >= 0 && n < (long long)E) ? x[n * CIN + ch] : 0.0f;
            }
            float sa = g[0] + g[1], da = fabsf(g[0] - g[1]);
            float sb = g[2] + g[3], db = fabsf(g[2] - g[3]);
            row[ch]           = xv;
            row[CIN + ch]     = sa + sb;
            row[2*CIN + ch]   = da + db;
            row[3*CIN + ch]   = fabsf(sa - sb);
            row[4*CIN + ch]   = fabsf(da - db);
        } else {
            row[ch]         = 0.0f;
            row[CIN + ch]   = 0.0f;
            row[2*CIN + ch] = 0.0f;
            row[3*CIN + ch] = 0.0f;
            row[4*CIN + ch] = 0.0f;
        }
    }
    __syncthreads();

    // ---- Stage 2: WMMA f32 GEMM: comb(128x160) @ Wl^T(160x64) ----
    // A 16x4 f32 layout: lanes 0-15 -> M=lane, VGPR0=K0,VGPR1=K1; lanes 16-31 -> K2/K3.
    // B 4x16 f32 layout: lanes 0-15 -> N=lane, VGPR0=K0,VGPR1=K1; lanes 16-31 -> K2/K3.
    const int half  = lane >> 4;   // 0 or 1
    const int lrow  = lane & 15;
    const int rbase = wave * 16;   // this wave's 16 rows of combined

    const float* arow = comb + (rbase + lrow) * STR + 2 * half;

    for (int nt = 0; nt < 4; ++nt) {
        v8f acc = {};
        const float* brow = Wl + (nt * 16 + lrow) * STR + 2 * half;
        #pragma unroll 8
        for (int ks = 0; ks < 40; ++ks) {
            v2f a  = *(const v2f*)(arow + ks * 4);   // 8B-aligned ds_load_b64
            v2f bf = *(const v2f*)(brow + ks * 4);
            acc = __builtin_amdgcn_wmma_f32_16x16x4_f32(
                /*neg_a=*/false, a, /*neg_b=*/false, bf,
                /*c_mod=*/(short)0, acc, /*reuse_a=*/false, /*reuse_b=*/false);
        }
        // C/D layout: VGPR v -> M = v + 8*half, col N = lrow
        const float bv = bias[nt * 16 + lrow];
        #pragma unroll
        for (int v = 0; v < 8; ++v) {
            int e = base + rbase + v + 8 * half;
            if (e < E) out[(long long)e * OUTF + nt * 16 + lrow] = acc[v] + bv;
        }
    }
}

extern "C" void kernel_launch(void* const* d_in, const int* in_sizes, int n_in,
                              void* d_out, int out_size, void* d_ws, size_t ws_size,
                              hipStream_t stream) {
    const float*     x    = (const float*)d_in[0];
    const long long* nbr  = (const long long*)d_in[1];
    const float*     W    = (const float*)d_in[2];
    const float*     bias = (const float*)d_in[3];
    float*           out  = (float*)d_out;

    const int E = in_sizes[0] / CIN;
    const int nblocks = (E + ROWS_PER_WG - 1) / ROWS_PER_WG;
    const size_t smem_bytes = (size_t)SMEM_FLOATS * sizeof(float);

    meshconv_wmma_f32_kernel<<<nblocks, NTHREADS, smem_bytes, stream>>>(
        x, nbr, W, bias, out, E);
}